// Model_76751065579687
// MI455X (gfx1250) — compile-verified
//
#include <hip/hip_runtime.h>
#include <math.h>

typedef __attribute__((ext_vector_type(2))) float v2f;
typedef __attribute__((ext_vector_type(8))) float v8f;

#define PA_EPS 1e-6f

static __device__ __forceinline__ v8f wmma4(v2f a, v2f b, v8f c) {
    // D = A(16x4,f32) * B(4x16,f32) + C(16x16,f32)
    return __builtin_amdgcn_wmma_f32_16x16x4_f32(false, a, false, b, (short)0, c,
                                                 false, false);
}

// ---------------------------------------------------------------------------
// Weight packing: Wp[Npad][Kpad] = zero-padded op(W); biasp[Npad] = padded bias.
//   wT_src=1: W is [N,K] row-major (torch linear weight)
//   wT_src=0: W is [K,N] row-major (ld = ldw_src)
// Runs once per launch per layer; makes all GEMM B/bias loads unconditional.
// ---------------------------------------------------------------------------
__global__ __launch_bounds__(256)
void pack_weight_kernel(const float* __restrict__ W, int ldw_src, int wT_src,
                        const float* __restrict__ bias,
                        float* __restrict__ Wp, float* __restrict__ biasp,
                        int N, int K, int Npad, int Kpad) {
    const int idx = blockIdx.x * blockDim.x + threadIdx.x;
    const int tot = Npad * Kpad;
    if (idx < tot) {
        const int n = idx / Kpad, k = idx % Kpad;
        float v = 0.f;
        if (n < N && k < K)
            v = wT_src ? W[(size_t)n * ldw_src + k] : W[(size_t)k * ldw_src + n];
        Wp[idx] = v;
    }
    if (idx < Npad && biasp != nullptr)
        biasp[idx] = (idx < N && bias != nullptr) ? bias[idx] : 0.f;
}

// ---------------------------------------------------------------------------
// Patch attention: out[b,c] = sum_s patches[b,c,s] * softmax_s(exp(-d_s)),
// d_s = sqrt(sum_c (x[b,c] - patch[b,c,s] + eps)^2).  One block per row,
// tile staged once in LDS and reused for both passes. Output row padded with
// zeros up to `padto` so downstream GEMMs can run guard-free.
// ---------------------------------------------------------------------------
__global__ __launch_bounds__(64)
void patch_attend_kernel(const float* __restrict__ X, const float* __restrict__ P,
                         float* __restrict__ Out, int C, int S, int ldo, int padto) {
    __shared__ float tile[63 * 49];
    __shared__ float xs[63];
    __shared__ float zz[49];
    __shared__ float pw[49];
    const int b = blockIdx.x, tid = threadIdx.x;
    const int CS = C * S;
    const float* pb = P + (size_t)b * CS;
    for (int i = tid; i < CS; i += 64) tile[i] = pb[i];
    for (int i = tid; i < C; i += 64) xs[i] = X[(size_t)b * C + i];
    __syncthreads();
    if (tid < S) {
        float acc = 0.f;
        for (int c = 0; c < C; ++c) {
            float d = xs[c] - tile[c * S + tid] + PA_EPS;
            acc += d * d;
        }
        zz[tid] = expf(-sqrtf(acc));
    }
    __syncthreads();
    if (tid == 0) {
        float mx = zz[0];
        for (int s = 1; s < S; ++s) mx = fmaxf(mx, zz[s]);
        float sum = 0.f;
        for (int s = 0; s < S; ++s) { pw[s] = expf(zz[s] - mx); sum += pw[s]; }
        float inv = 1.f / sum;
        for (int s = 0; s < S; ++s) pw[s] *= inv;
    }
    __syncthreads();
    for (int c = tid; c < padto; c += 64) {
        float acc = 0.f;
        if (c < C)
            for (int s = 0; s < S; ++s) acc += tile[c * S + s] * pw[s];
        Out[(size_t)b * ldo + c] = acc;
    }
}

// ---------------------------------------------------------------------------
// Guard-free wave-tile WMMA GEMM on padded operands:
//   Out[M,Npad] = X[M,Kpad] @ Wp^T (+biasp), Wp is packed [Npad][Kpad].
// One wave per 16x16 tile; inner loop = 2x global_load_b64 + v_wmma.
// Optional second (padded) destination for producing both an exact-stride
// output and a zero-padded workspace copy in one pass.
// ---------------------------------------------------------------------------
__global__ __launch_bounds__(32)
void gemm_wmma_kernel(const float* __restrict__ X, int ldx,
                      const float* __restrict__ Wp, int Kpad,
                      const float* __restrict__ biasp,
                      float* __restrict__ Out, int ldo, int Nstore,
                      float* __restrict__ Out2, int ldo2, int Nstore2) {
    const int mtile = blockIdx.x, ntile = blockIdx.y;
    const int lane = threadIdx.x;
    const int half = lane >> 4, l16 = lane & 15;
    const int koff = half * 2;
    const int col = ntile * 16 + l16;
    const float* xp = X + (size_t)(mtile * 16 + l16) * ldx + koff;
    const float* wp = Wp + (size_t)col * Kpad + koff;

    v8f acc = {};
    for (int kb = 0; kb < Kpad; kb += 4) {
        v2f a, bb;
        a.x = xp[0]; a.y = xp[1];
        bb.x = wp[0]; bb.y = wp[1];
        acc = wmma4(a, bb, acc);
        xp += 4; wp += 4;
    }
    const float bv = (biasp != nullptr) ? biasp[col] : 0.f;
    if (col < Nstore) {
#pragma unroll
        for (int r = 0; r < 8; ++r)
            Out[(size_t)(mtile * 16 + r + half * 8) * ldo + col] = acc[r] + bv;
    }
    if (Out2 != nullptr && col < Nstore2) {
#pragma unroll
        for (int r = 0; r < 8; ++r)
            Out2[(size_t)(mtile * 16 + r + half * 8) * ldo2 + col] = acc[r] + bv;
    }
}

// ---------------------------------------------------------------------------
// Fused attention scores + softmax:
//   S = H1[B,64(pad of 60)] @ D2[64,64(pad of 60)]^T ; A = softmax_rows(S)
// One wave per 16 rows; 4 N-tile accumulators; in-wave row softmax via
// shfl_xor (masks 1..8 stay inside the 16-lane half that owns each row).
// ---------------------------------------------------------------------------
__global__ __launch_bounds__(32)
void attn_softmax_kernel(const float* __restrict__ H1, const float* __restrict__ D2,
                         float* __restrict__ A) {
    const int mtile = blockIdx.x;
    const int lane = threadIdx.x;
    const int half = lane >> 4, l16 = lane & 15;
    const int koff = half * 2;
    const float* xrow = H1 + (size_t)(mtile * 16 + l16) * 64 + koff;
    const float* drow = D2 + (size_t)l16 * 64 + koff;

    v8f acc[4] = {{}, {}, {}, {}};
    for (int kb = 0; kb < 64; kb += 4) {
        v2f a; a.x = xrow[kb]; a.y = xrow[kb + 1];
#pragma unroll
        for (int t = 0; t < 4; ++t) {
            v2f bb;
            bb.x = drow[(size_t)t * 16 * 64 + kb];
            bb.y = drow[(size_t)t * 16 * 64 + kb + 1];
            acc[t] = wmma4(a, bb, acc[t]);
        }
    }
#pragma unroll
    for (int r = 0; r < 8; ++r) {
        float mx = fmaxf(fmaxf(acc[0][r], acc[1][r]), fmaxf(acc[2][r], acc[3][r]));
        for (int msk = 1; msk <= 8; msk <<= 1) mx = fmaxf(mx, __shfl_xor(mx, msk, 32));
        float ev[4], sum = 0.f;
#pragma unroll
        for (int t = 0; t < 4; ++t) { ev[t] = expf(acc[t][r] - mx); sum += ev[t]; }
        for (int msk = 1; msk <= 8; msk <<= 1) sum += __shfl_xor(sum, msk, 32);
        const float inv = 1.f / sum;
        const int row = mtile * 16 + r + half * 8;
#pragma unroll
        for (int t = 0; t < 4; ++t) A[(size_t)row * 64 + t * 16 + l16] = ev[t] * inv;
    }
}

// ---------------------------------------------------------------------------
// LiDAR branch MLP (K=1 input): 1 -> 15 -> 30. Pure VALU; WMMA would waste
// 3/4 of each fragment at K=1. Writes exact output and optional padded copy.
// ---------------------------------------------------------------------------
__global__ __launch_bounds__(256)
void lidar_mlp_kernel(const float* __restrict__ sp,
                      const float* __restrict__ w21, const float* __restrict__ b21,
                      const float* __restrict__ w22, const float* __restrict__ b22,
                      float* __restrict__ out, int ldo, int padto,
                      float* __restrict__ out2, int ldo2, int padto2, int B) {
    const int t = blockIdx.x * blockDim.x + threadIdx.x;
    if (t >= B) return;
    const float x = sp[t];
    float h[15];
#pragma unroll
    for (int j = 0; j < 15; ++j) h[j] = x * w21[j] + b21[j];
#pragma unroll
    for (int c = 0; c < 30; ++c) {
        float a = b22[c];
#pragma unroll
        for (int j = 0; j < 15; ++j) a += h[j] * w22[c * 15 + j];
        out[(size_t)t * ldo + c] = a;
        if (out2 != nullptr) out2[(size_t)t * ldo2 + c] = a;
    }
    for (int c = 30; c < padto; ++c) out[(size_t)t * ldo + c] = 0.f;
    if (out2 != nullptr)
        for (int c = 30; c < padto2; ++c) out2[(size_t)t * ldo2 + c] = 0.f;
}

// ---------------------------------------------------------------------------
// Fused head: mid = concat(LiDAR_A, HSI_A)[B,120] @ wo1^T + bo1  -> LDS
//             out = mid[B,60] @ wo2^T + bo2 -> [B,7]
// Uses packed wo1 [64][120] / wo2 [16][60] so all loads are unconditional.
// ---------------------------------------------------------------------------
__global__ __launch_bounds__(32)
void final_head_kernel(const float* __restrict__ LA, const float* __restrict__ HA,
                       const float* __restrict__ Wpo1, const float* __restrict__ bpo1,
                       const float* __restrict__ Wpo2, const float* __restrict__ bpo2,
                       float* __restrict__ Out) {
    __shared__ float smid[16][68];
    const int mtile = blockIdx.x;
    const int lane = threadIdx.x;
    const int half = lane >> 4, l16 = lane & 15;
    const int koff = half * 2;
    const float* larow = LA + (size_t)(mtile * 16 + l16) * 60 + koff;
    const float* harow = HA + (size_t)(mtile * 16 + l16) * 60 + koff;
    const float* w1p = Wpo1 + (size_t)l16 * 120 + koff;

    v8f acc[4] = {{}, {}, {}, {}};
#pragma unroll
    for (int kb = 0; kb < 120; kb += 4) {
        v2f a;
        if (kb < 60) { a.x = larow[kb]; a.y = larow[kb + 1]; }
        else         { a.x = harow[kb - 60]; a.y = harow[kb - 60 + 1]; }
#pragma unroll
        for (int t = 0; t < 4; ++t) {
            v2f bb;
            bb.x = w1p[(size_t)t * 16 * 120 + kb];
            bb.y = w1p[(size_t)t * 16 * 120 + kb + 1];
            acc[t] = wmma4(a, bb, acc[t]);
        }
    }
#pragma unroll
    for (int t = 0; t < 4; ++t) {
        const int col = t * 16 + l16;
        const float bv = bpo1[col];
#pragma unroll
        for (int r = 0; r < 8; ++r) smid[r + half * 8][col] = acc[t][r] + bv;
    }
    __syncthreads();

    v8f o = {};
    const float* w2p = Wpo2 + (size_t)l16 * 60 + koff;
#pragma unroll
    for (int kb = 0; kb < 60; kb += 4) {
        v2f a; a.x = smid[l16][kb + koff]; a.y = smid[l16][kb + koff + 1];
        v2f bb; bb.x = w2p[kb]; bb.y = w2p[kb + 1];
        o = wmma4(a, bb, o);
    }
    if (l16 < 7) {
        const float bv = bpo2[l16];
#pragma unroll
        for (int r = 0; r < 8; ++r)
            Out[(size_t)(mtile * 16 + r + half * 8) * 7 + l16] = o[r] + bv;
    }
}

// ---------------------------------------------------------------------------
// newD partials: per-block accumulation of data^T @ A over a row slice,
// deterministic fixed-order finalize afterwards (no fp atomics).
// data is the padded [B,32] copy.
// ---------------------------------------------------------------------------
__global__ __launch_bounds__(256)
void newD_partial_kernel(const float* __restrict__ data, const float* __restrict__ A,
                         float* __restrict__ partial, int B) {
    __shared__ float sd[64 * 32];
    __shared__ float sa[64 * 64];
    const int blk = blockIdx.x, tid = threadIdx.x;
    const int rowsPerBlk = B / gridDim.x;
    const int r0 = blk * rowsPerBlk;
    float acc[8];
#pragma unroll
    for (int i = 0; i < 8; ++i) acc[i] = 0.f;
    for (int ch = 0; ch < rowsPerBlk; ch += 64) {
        const int base = r0 + ch;
        for (int i = tid; i < 64 * 32; i += 256) sd[i] = data[(size_t)base * 32 + i];
        for (int i = tid; i < 64 * 64; i += 256) sa[i] = A[(size_t)base * 64 + i];
        __syncthreads();
#pragma unroll
        for (int i = 0; i < 8; ++i) {
            const int e = tid + i * 256;
            if (e < 1920) {
                const int c = e >> 6, j = e & 63;
                float s = 0.f;
                for (int r = 0; r < 64; ++r) s += sd[r * 32 + c] * sa[r * 64 + j];
                acc[i] += s;
            }
        }
        __syncthreads();
    }
#pragma unroll
    for (int i = 0; i < 8; ++i) {
        const int e = tid + i * 256;
        if (e < 1920) partial[(size_t)blk * 1920 + e] = acc[i];
    }
}

__global__ __launch_bounds__(256)
void newD_finalize_kernel(const float* __restrict__ Dpad /*[64,32]*/,
                          const float* __restrict__ partial,
                          const int* __restrict__ m_ptr, float* __restrict__ newD,
                          int nblk) {
    const int e = blockIdx.x * blockDim.x + threadIdx.x;
    if (e >= 1920) return;
    const int c = e >> 6, j = e & 63;
    float s = 0.f;
    for (int b = 0; b < nblk; ++b) s += partial[(size_t)b * 1920 + e];
    const float mf = (float)m_ptr[0];
    newD[e] = mf * Dpad[(size_t)j * 32 + c] + (1.f - mf) * s;
}

// ---------------------------------------------------------------------------
extern "C" void kernel_launch(void* const* d_in, const int* in_sizes, int n_in,
                              void* d_out, int out_size, void* d_ws, size_t ws_size,
                              hipStream_t stream) {
    (void)n_in; (void)out_size; (void)ws_size;
    const float* H_HSI       = (const float*)d_in[0];
    const float* H_LiDAR     = (const float*)d_in[1];
    const float* HSI_D       = (const float*)d_in[2];
    const float* LiDAR_D     = (const float*)d_in[3];
    const float* Patch_HSI   = (const float*)d_in[4];
    const float* Patch_LiDAR = (const float*)d_in[5];
    const float* D_Patch_HSI   = (const float*)d_in[6];
    const float* D_Patch_LiDAR = (const float*)d_in[7];
    const int*   m_ptr = (const int*)d_in[9];
    const float* w1_1 = (const float*)d_in[12]; const float* b1_1 = (const float*)d_in[13];
    const float* w1_2 = (const float*)d_in[14]; const float* b1_2 = (const float*)d_in[15];
    const float* w2_1 = (const float*)d_in[16]; const float* b2_1 = (const float*)d_in[17];
    const float* w2_2 = (const float*)d_in[18]; const float* b2_2 = (const float*)d_in[19];
    const float* wa1  = (const float*)d_in[20]; const float* ba1  = (const float*)d_in[21];
    const float* wa2  = (const float*)d_in[22]; const float* ba2  = (const float*)d_in[23];
    const float* wo1  = (const float*)d_in[24]; const float* bo1  = (const float*)d_in[25];
    const float* wo2  = (const float*)d_in[26]; const float* bo2  = (const float*)d_in[27];

    const int B  = in_sizes[0] / 63;   // 16384
    const int Dn = in_sizes[2] / 63;   // 64

    // d_out layout (reference tuple order)
    float* out_data_HSI   = (float*)d_out;                      // [B,30]
    float* out_data_LiDAR = out_data_HSI + (size_t)B * 30;      // [B,30]
    float* out_HSI_A      = out_data_LiDAR + (size_t)B * 30;    // [B,60]
    float* out_HSI_Dn     = out_HSI_A + (size_t)B * 60;         // [30,Dn]
    float* out_LiDAR_A    = out_HSI_Dn + (size_t)30 * Dn;       // [B,60]
    float* out_LiDAR_Dn   = out_LiDAR_A + (size_t)B * 60;       // [30,Dn]
    float* out_final      = out_LiDAR_Dn + (size_t)30 * Dn;     // [B,7]

    // workspace layout (all activation buffers K-padded for guard-free GEMM)
    float* w = (float*)d_ws;
    float* sp_hsi    = w;  w += (size_t)B * 64;    // [B,64]  (63 + pad)
    float* sp_lidar  = w;  w += (size_t)B;         // [B,1]
    float* dsp_hsi   = w;  w += (size_t)Dn * 64;   // [Dn,64]
    float* dsp_lidar = w;  w += (size_t)Dn;        // [Dn,1]
    float* h1a       = w;  w += (size_t)B * 48;    // [B,48]  (45 + pad)
    float* h1d       = w;  w += (size_t)Dn * 48;
    float* dat_h_pad = w;  w += (size_t)B * 32;    // padded copy of data_HSI
    float* dat_l_pad = w;  w += (size_t)B * 32;    // padded copy of data_LiDAR
    float* Dh        = w;  w += (size_t)Dn * 32;   // [Dn,32] (30 + pad)
    float* Dl        = w;  w += (size_t)Dn * 32;
    float* H1h       = w;  w += (size_t)B * 64;    // [B,64] (60 + pad)
    float* H1l       = w;  w += (size_t)B * 64;
    float* D2h       = w;  w += (size_t)Dn * 64;
    float* D2l       = w;  w += (size_t)Dn * 64;
    float* Ah        = w;  w += (size_t)B * 64;
    float* Al        = w;  w += (size_t)B * 64;
    float* outh      = w;  w += (size_t)B * 32;    // [B,32] (30 + pad)
    float* outl      = w;  w += (size_t)B * 32;
    float* part_h    = w;  w += (size_t)64 * 1920;
    float* part_l    = w;  w += (size_t)64 * 1920;
    // packed weights + biases
    float* Wp11   = w;  w += 48 * 64;    float* bp11  = w;  w += 48;
    float* Wp12   = w;  w += 32 * 48;    float* bp12  = w;  w += 32;
    float* Wpa1   = w;  w += 64 * 32;    float* bpa1  = w;  w += 64;
    float* Wpa2   = w;  w += 64 * 32;    float* bpa2  = w;  w += 64;
    float* Wpo1   = w;  w += 64 * 120;   float* bpo1  = w;  w += 64;
    float* Wpo2   = w;  w += 16 * 60;    float* bpo2  = w;  w += 16;
    float* WpDh   = w;  w += 32 * 64;
    float* WpDl   = w;  w += 32 * 64;

    const int MT = B / 16;   // 1024 row-tiles
    const int DT = Dn / 16;  // 4 row-tiles

    // ---- pack static weights (tiny) ----
    pack_weight_kernel<<<12, 256, 0, stream>>>(w1_1, 63, 1, b1_1, Wp11, bp11, 45, 63, 48, 64);
    pack_weight_kernel<<<6,  256, 0, stream>>>(w1_2, 45, 1, b1_2, Wp12, bp12, 30, 45, 32, 48);
    pack_weight_kernel<<<8,  256, 0, stream>>>(wa1, 30, 1, ba1, Wpa1, bpa1, 60, 30, 64, 32);
    pack_weight_kernel<<<8,  256, 0, stream>>>(wa2, 30, 1, ba2, Wpa2, bpa2, 60, 30, 64, 32);
    pack_weight_kernel<<<30, 256, 0, stream>>>(wo1, 120, 1, bo1, Wpo1, bpo1, 60, 120, 64, 120);
    pack_weight_kernel<<<4,  256, 0, stream>>>(wo2, 60, 1, bo2, Wpo2, bpo2, 7, 60, 16, 60);

    // ---- patch attention (memory-bound stage) ----
    patch_attend_kernel<<<B, 64, 0, stream>>>(H_HSI, Patch_HSI, sp_hsi, 63, 49, 64, 64);
    patch_attend_kernel<<<B, 64, 0, stream>>>(H_LiDAR, Patch_LiDAR, sp_lidar, 1, 49, 1, 1);
    patch_attend_kernel<<<Dn, 64, 0, stream>>>(HSI_D, D_Patch_HSI, dsp_hsi, 63, 49, 64, 64);
    patch_attend_kernel<<<Dn, 64, 0, stream>>>(LiDAR_D, D_Patch_LiDAR, dsp_lidar, 1, 49, 1, 1);

    // ---- lin1 chain: 63 -> 45 -> 30 (WMMA f32) ----
    gemm_wmma_kernel<<<dim3(MT, 3), 32, 0, stream>>>(sp_hsi, 64, Wp11, 64, bp11,
                                                     h1a, 48, 48, nullptr, 0, 0);
    gemm_wmma_kernel<<<dim3(MT, 2), 32, 0, stream>>>(h1a, 48, Wp12, 48, bp12,
                                                     out_data_HSI, 30, 30,
                                                     dat_h_pad, 32, 32);
    gemm_wmma_kernel<<<dim3(DT, 3), 32, 0, stream>>>(dsp_hsi, 64, Wp11, 64, bp11,
                                                     h1d, 48, 48, nullptr, 0, 0);
    gemm_wmma_kernel<<<dim3(DT, 2), 32, 0, stream>>>(h1d, 48, Wp12, 48, bp12,
                                                     Dh, 32, 32, nullptr, 0, 0);

    // ---- lin2 chain (K=1): VALU ----
    lidar_mlp_kernel<<<(B + 255) / 256, 256, 0, stream>>>(
        sp_lidar, w2_1, b2_1, w2_2, b2_2,
        out_data_LiDAR, 30, 30, dat_l_pad, 32, 32, B);
    lidar_mlp_kernel<<<1, 256, 0, stream>>>(
        dsp_lidar, w2_1, b2_1, w2_2, b2_2, Dl, 32, 32, nullptr, 0, 0, Dn);

    // ---- pack runtime dictionary matrices as B operands: Wp[n][k]=D[k,n] ----
    pack_weight_kernel<<<8, 256, 0, stream>>>(Dh, 32, 0, nullptr, WpDh, nullptr, 30, 64, 32, 64);
    pack_weight_kernel<<<8, 256, 0, stream>>>(Dl, 32, 0, nullptr, WpDl, nullptr, 30, 64, 32, 64);

    // ---- attention projections: 30 -> 60 (padded to 64 with zeros) ----
    gemm_wmma_kernel<<<dim3(MT, 4), 32, 0, stream>>>(dat_h_pad, 32, Wpa1, 32, bpa1,
                                                     H1h, 64, 64, nullptr, 0, 0);
    gemm_wmma_kernel<<<dim3(DT, 4), 32, 0, stream>>>(Dh, 32, Wpa2, 32, bpa2,
                                                     D2h, 64, 64, nullptr, 0, 0);
    gemm_wmma_kernel<<<dim3(MT, 4), 32, 0, stream>>>(dat_l_pad, 32, Wpa1, 32, bpa1,
                                                     H1l, 64, 64, nullptr, 0, 0);
    gemm_wmma_kernel<<<dim3(DT, 4), 32, 0, stream>>>(Dl, 32, Wpa2, 32, bpa2,
                                                     D2l, 64, 64, nullptr, 0, 0);

    // ---- scores + softmax (fused, in-wave) ----
    attn_softmax_kernel<<<MT, 32, 0, stream>>>(H1h, D2h, Ah);
    attn_softmax_kernel<<<MT, 32, 0, stream>>>(H1l, D2l, Al);

    // ---- out = A @ D^T (K=64, N=30, no bias) ----
    gemm_wmma_kernel<<<dim3(MT, 2), 32, 0, stream>>>(Ah, 64, WpDh, 64, nullptr,
                                                     outh, 32, 32, nullptr, 0, 0);
    gemm_wmma_kernel<<<dim3(MT, 2), 32, 0, stream>>>(Al, 64, WpDl, 64, nullptr,
                                                     outl, 32, 32, nullptr, 0, 0);

    // ---- feature projections: 30 -> 60 into d_out (exact stride 60) ----
    gemm_wmma_kernel<<<dim3(MT, 4), 32, 0, stream>>>(outh, 32, Wpa1, 32, bpa1,
                                                     out_HSI_A, 60, 60, nullptr, 0, 0);
    gemm_wmma_kernel<<<dim3(MT, 4), 32, 0, stream>>>(outl, 32, Wpa1, 32, bpa1,
                                                     out_LiDAR_A, 60, 60, nullptr, 0, 0);

    // ---- fused classification head: 120 -> 60 -> 7 ----
    final_head_kernel<<<MT, 32, 0, stream>>>(out_LiDAR_A, out_HSI_A,
                                             Wpo1, bpo1, Wpo2, bpo2, out_final);

    // ---- momentum dictionary update (deterministic 2-pass reduction) ----
    newD_partial_kernel<<<64, 256, 0, stream>>>(dat_h_pad, Ah, part_h, B);
    newD_finalize_kernel<<<8, 256, 0, stream>>>(Dh, part_h, m_ptr, out_HSI_Dn, 64);
    newD_partial_kernel<<<64, 256, 0, stream>>>(dat_l_pad, Al, part_l, B);
    newD_finalize_kernel<<<8, 256, 0, stream>>>(Dl, part_l, m_ptr, out_LiDAR_Dn, 64);
}